// MIVSPool_1194000908389
// MI455X (gfx1250) — compile-verified
//
#include <hip/hip_runtime.h>

typedef float v2f __attribute__((ext_vector_type(2)));
typedef float v8f __attribute__((ext_vector_type(8)));

#define MIS_MAX_ITERS 48

// ---------------------------------------------------------------------------
// Init: zero x_pool region, batch_out = -1, MIS state, iteration flags.
// ---------------------------------------------------------------------------
__global__ void mivs_init(float* __restrict__ out, long nc, long batchOff,
                          unsigned long long* __restrict__ vmax,
                          unsigned long long* __restrict__ best64,
                          unsigned int* __restrict__ p,
                          unsigned int* __restrict__ q,
                          unsigned int* __restrict__ hasNbr,
                          int* __restrict__ flags, int N) {
  long stride = (long)gridDim.x * blockDim.x;
  long tid = (long)blockIdx.x * blockDim.x + threadIdx.x;
  for (long i = tid; i < nc; i += stride) out[i] = 0.0f;  // x_pool
  for (long i = tid; i < N; i += stride) {
    out[batchOff + i] = -1.0f;
    p[i] = 0u; q[i] = 1u; hasNbr[i] = 0u;
    vmax[i] = 0ull; best64[i] = 0ull;
  }
  if (tid == 0) {
    flags[0] = 1;
    for (int j = 1; j <= MIS_MAX_ITERS; ++j) flags[j] = 0;
  }
}

// ---------------------------------------------------------------------------
// score = sigmoid(x @ w)  via V_WMMA_F32_16X16X4_F32 (exact fp32).
// One wave handles 16 rows. A-fragment per ISA layout: VGPR0 = K0 (lanes
// 0-15) / K2 (lanes 16-31), VGPR1 = K1/K3 -> contiguous 8B pair per lane.
// B = w slice replicated across all 16 columns, so every column of D holds
// the dot product; lanes 0/16 extract column 0 per the C/D VGPR layout.
// Also emits key64 = (bits(score) << 32) | node  (monotone MIS rank key).
// ---------------------------------------------------------------------------
__global__ void mivs_score_wmma(const float* __restrict__ x,
                                const float* __restrict__ w,
                                float* __restrict__ score,
                                unsigned long long* __restrict__ key64,
                                int N, int C) {
  int gtid = blockIdx.x * blockDim.x + threadIdx.x;
  int wave = gtid >> 5;
  int lane = threadIdx.x & 31;
  int rowBase = wave * 16;
  if (rowBase >= N) return;                 // uniform per wave
  int m    = lane & 15;
  int half = lane >> 4;                     // 0: lanes 0-15, 1: lanes 16-31
  int koff = half ? 2 : 0;

  v8f acc = {0.f, 0.f, 0.f, 0.f, 0.f, 0.f, 0.f, 0.f};

  if (rowBase + 16 <= N) {
    // Fast path: full tile, branchless -> global_load_b64 + v_wmma loop.
    const float* __restrict__ xrow = x + (long)(rowBase + m) * C + koff;
    const float* __restrict__ wk   = w + koff;
#pragma unroll 4
    for (int k0 = 0; k0 < C; k0 += 4) {
      v2f a = *(const v2f*)(xrow + k0);     // {x[r][k], x[r][k+1]}
      v2f b = *(const v2f*)(wk + k0);       // {w[k], w[k+1]}
      acc = __builtin_amdgcn_wmma_f32_16x16x4_f32(
          false, a, false, b, (short)0, acc, false, false);
    }
  } else {
    // Tail tile (at most one wave): guarded loads.
    int row  = rowBase + m;
    bool rowOk = row < N;
    for (int k0 = 0; k0 < C; k0 += 4) {
      int ka = k0 + koff;
      v2f a, b;
      a.x = rowOk ? x[(long)row * C + ka]     : 0.0f;
      a.y = rowOk ? x[(long)row * C + ka + 1] : 0.0f;
      b.x = w[ka];
      b.y = w[ka + 1];
      acc = __builtin_amdgcn_wmma_f32_16x16x4_f32(
          false, a, false, b, (short)0, acc, false, false);
    }
  }

  // D layout: VGPR r -> M=r (lanes 0-15) / M=8+r (lanes 16-31), N=lane%16.
  if ((lane & 15) == 0) {
    int base = rowBase + (half ? 8 : 0);
#pragma unroll
    for (int r = 0; r < 8; ++r) {
      int rr = base + r;
      if (rr < N) {
        float s = 1.0f / (1.0f + __expf(-acc[r]));
        score[rr] = s;
        // s > 0 always => raw float bits are order-preserving.
        key64[rr] = ((unsigned long long)__float_as_uint(s) << 32) |
                    (unsigned int)rr;
      }
    }
  }
}

// ---------------------------------------------------------------------------
// MIS round kernels (device-flag chained; early-exit when converged).
// ---------------------------------------------------------------------------
__global__ void mis_vmax(const int* __restrict__ row, const int* __restrict__ col,
                         const unsigned long long* __restrict__ key,
                         const unsigned int* __restrict__ q,
                         unsigned long long* __restrict__ vmax,
                         const int* __restrict__ flags, int it, int E) {
  if (flags[it] == 0) return;
  int e = blockIdx.x * blockDim.x + threadIdx.x;
  if (e >= E) return;
  int v = row[e], u = col[e];
  if (v == u) return;                       // remove_self_loops
  if (q[u]) atomicMax(&vmax[v], key[u]);
}

__global__ void mis_survive(const unsigned long long* __restrict__ key,
                            const unsigned long long* __restrict__ vmax,
                            unsigned int* __restrict__ p,
                            const unsigned int* __restrict__ q,
                            const int* __restrict__ flags, int it, int N) {
  if (flags[it] == 0) return;
  int v = blockIdx.x * blockDim.x + threadIdx.x;
  if (v >= N) return;
  if (q[v] && (vmax[v] < key[v])) p[v] = 1u;   // local max among active
}

__global__ void mis_nbr(const int* __restrict__ row, const int* __restrict__ col,
                        const unsigned int* __restrict__ p,
                        unsigned int* __restrict__ hasNbr,
                        const int* __restrict__ flags, int it, int E) {
  if (flags[it] == 0) return;
  int e = blockIdx.x * blockDim.x + threadIdx.x;
  if (e >= E) return;
  int v = row[e], u = col[e];
  if (v != u && p[u]) hasNbr[v] = 1u;        // benign race: all write 1
}

__global__ void mis_update(unsigned int* __restrict__ q,
                           const unsigned int* __restrict__ p,
                           unsigned int* __restrict__ hasNbr,
                           unsigned long long* __restrict__ vmax,
                           int* __restrict__ flags, int it, int N) {
  if (flags[it] == 0) return;
  int v = blockIdx.x * blockDim.x + threadIdx.x;
  if (v >= N) return;
  unsigned int hn = hasNbr[v];
  hasNbr[v] = 0u;                            // reset for next round
  vmax[v] = 0ull;
  unsigned int qq = (hn == 0u) && (p[v] == 0u);
  q[v] = qq ? 1u : 0u;
  if (qq) flags[it + 1] = 1;                 // benign race: all write 1
}

// ---------------------------------------------------------------------------
// Assignment: each node -> surviving neighbor with max (score, col).
// ---------------------------------------------------------------------------
__global__ void assign_edge(const int* __restrict__ row, const int* __restrict__ col,
                            const unsigned int* __restrict__ p,
                            const float* __restrict__ score,
                            unsigned long long* __restrict__ best64, int E) {
  int e = blockIdx.x * blockDim.x + threadIdx.x;
  if (e >= E) return;
  int v = row[e], u = col[e];
  if (v != u && p[u]) {
    unsigned long long k =
        ((unsigned long long)__float_as_uint(score[u]) << 32) | (unsigned int)u;
    atomicMax(&best64[v], k);
  }
}

__global__ void assign_node(const unsigned int* __restrict__ p,
                            const unsigned long long* __restrict__ best64,
                            int* __restrict__ assigned, int N) {
  int v = blockIdx.x * blockDim.x + threadIdx.x;
  if (v >= N) return;
  assigned[v] = p[v] ? v : (int)(best64[v] & 0xFFFFFFFFull);
}

// ---------------------------------------------------------------------------
// Exclusive prefix sum of p  -> new_id (compacted cluster ids).
// ---------------------------------------------------------------------------
__global__ void scan_block(const unsigned int* __restrict__ p,
                           int* __restrict__ blockSums, int N) {
  __shared__ int sh[512];
  int v = blockIdx.x * 512 + threadIdx.x;
  sh[threadIdx.x] = (v < N) ? (int)p[v] : 0;
  __syncthreads();
  for (int s = 256; s > 0; s >>= 1) {
    if ((int)threadIdx.x < s) sh[threadIdx.x] += sh[threadIdx.x + s];
    __syncthreads();
  }
  if (threadIdx.x == 0) blockSums[blockIdx.x] = sh[0];
}

__global__ void scan_sums(int* __restrict__ blockSums, int NB) {
  if (blockIdx.x == 0 && threadIdx.x == 0) {
    int acc = 0;
    for (int i = 0; i < NB; ++i) { int t = blockSums[i]; blockSums[i] = acc; acc += t; }
  }
}

__global__ void scan_final(const unsigned int* __restrict__ p,
                           const int* __restrict__ blockSums,
                           int* __restrict__ newid, int N) {
  __shared__ int sh[512];
  int v = blockIdx.x * 512 + threadIdx.x;
  int val = (v < N) ? (int)p[v] : 0;
  sh[threadIdx.x] = val;
  __syncthreads();
  for (int s = 1; s < 512; s <<= 1) {
    int t = ((int)threadIdx.x >= s) ? sh[threadIdx.x - s] : 0;
    __syncthreads();
    sh[threadIdx.x] += t;
    __syncthreads();
  }
  if (v < N) newid[v] = blockSums[blockIdx.x] + sh[threadIdx.x] - val;  // exclusive
}

// ---------------------------------------------------------------------------
// Finalize nodes: colS, p output, batch_out (survivor slots).
// ---------------------------------------------------------------------------
__global__ void finalize_nodes(const unsigned int* __restrict__ p,
                               const int* __restrict__ assigned,
                               const int* __restrict__ newid,
                               const int* __restrict__ batch,
                               int* __restrict__ colS,
                               float* __restrict__ out,
                               long batchOff, long pOff, int N) {
  int v = blockIdx.x * blockDim.x + threadIdx.x;
  if (v >= N) return;
  colS[v] = newid[assigned[v]];
  out[pOff + v] = p[v] ? 1.0f : 0.0f;
  if (p[v]) out[batchOff + newid[v]] = (float)batch[v];
}

// x_pool[colS[v], :] += score[v] * x[v, :]   (N*C < 2^31 -> 32-bit index)
__global__ void pool_scatter(const float* __restrict__ x,
                             const float* __restrict__ score,
                             const int* __restrict__ colS,
                             float* __restrict__ out_xpool,
                             unsigned int nc, int C) {
  unsigned int i = blockIdx.x * blockDim.x + threadIdx.x;
  if (i >= nc) return;
  unsigned int v = i / (unsigned int)C;
  unsigned int c = i - v * (unsigned int)C;
  atomicAdd(&out_xpool[(long)colS[v] * C + c], x[i] * score[v]);
}

__global__ void finalize_edges(const int* __restrict__ row, const int* __restrict__ col,
                               const float* __restrict__ w,
                               const int* __restrict__ colS,
                               float* __restrict__ out,
                               long rowOff, long colOff, long wOff, int E) {
  int e = blockIdx.x * blockDim.x + threadIdx.x;
  if (e >= E) return;
  out[rowOff + e] = (float)colS[row[e]];
  out[colOff + e] = (float)colS[col[e]];
  out[wOff + e]   = w[e];
}

// ---------------------------------------------------------------------------
extern "C" void kernel_launch(void* const* d_in, const int* in_sizes, int n_in,
                              void* d_out, int out_size, void* d_ws, size_t ws_size,
                              hipStream_t stream) {
  const float* x          = (const float*)d_in[0];
  const int*   edge_index = (const int*)d_in[1];
  const float* ew         = (const float*)d_in[2];
  const int*   batch      = (const int*)d_in[3];
  const float* w_score    = (const float*)d_in[4];

  const int E = in_sizes[2];
  const int N = in_sizes[3];
  const int C = in_sizes[4];
  const int* row = edge_index;
  const int* col = edge_index + E;

  // Workspace carve-up (8B-aligned arrays first). ~5.3 MB total.
  char* ws = (char*)d_ws;
  unsigned long long* key64  = (unsigned long long*)(ws);
  unsigned long long* vmax   = (unsigned long long*)(ws + (size_t)N * 8);
  unsigned long long* best64 = (unsigned long long*)(ws + (size_t)N * 16);
  float*        score   = (float*)(ws + (size_t)N * 24);
  unsigned int* p       = (unsigned int*)(ws + (size_t)N * 28);
  unsigned int* q       = (unsigned int*)(ws + (size_t)N * 32);
  unsigned int* hasNbr  = (unsigned int*)(ws + (size_t)N * 36);
  int*          assigned= (int*)(ws + (size_t)N * 40);
  int*          newid   = (int*)(ws + (size_t)N * 44);
  int*          colS    = (int*)(ws + (size_t)N * 48);
  int*          blockSums = (int*)(ws + (size_t)N * 52);
  int*          flags     = blockSums + 1024;

  float* out = (float*)d_out;
  const long nc       = (long)N * C;
  const long rowOff   = nc;
  const long colOff   = rowOff + E;
  const long wOff     = colOff + E;
  const long batchOff = wOff + E;
  const long pOff     = batchOff + N;

  const int eblocks = (E + 255) / 256;
  const int nblocks = (N + 255) / 256;

  mivs_init<<<1024, 256, 0, stream>>>(out, nc, batchOff, vmax, best64, p, q,
                                      hasNbr, flags, N);

  // score via WMMA: one wave per 16 rows.
  {
    long waves = ((long)N + 15) / 16;
    long threads = waves * 32;
    int blocks = (int)((threads + 255) / 256);
    mivs_score_wmma<<<blocks, 256, 0, stream>>>(x, w_score, score, key64, N, C);
  }

  // Fixed-round Luby MIS with device-side convergence flags.
  for (int it = 0; it < MIS_MAX_ITERS; ++it) {
    mis_vmax   <<<eblocks, 256, 0, stream>>>(row, col, key64, q, vmax, flags, it, E);
    mis_survive<<<nblocks, 256, 0, stream>>>(key64, vmax, p, q, flags, it, N);
    mis_nbr    <<<eblocks, 256, 0, stream>>>(row, col, p, hasNbr, flags, it, E);
    mis_update <<<nblocks, 256, 0, stream>>>(q, p, hasNbr, vmax, flags, it, N);
  }

  // Assignment to best surviving neighbor.
  assign_edge<<<eblocks, 256, 0, stream>>>(row, col, p, score, best64, E);
  assign_node<<<nblocks, 256, 0, stream>>>(p, best64, assigned, N);

  // new_id = exclusive prefix sum of p.
  const int NB = (N + 511) / 512;
  scan_block<<<NB, 512, 0, stream>>>(p, blockSums, N);
  scan_sums <<<1, 32, 0, stream>>>(blockSums, NB);
  scan_final<<<NB, 512, 0, stream>>>(p, blockSums, newid, N);

  // Outputs.
  finalize_nodes<<<nblocks, 256, 0, stream>>>(p, assigned, newid, batch, colS,
                                              out, batchOff, pOff, N);
  pool_scatter<<<(int)((nc + 255) / 256), 256, 0, stream>>>(
      x, score, colS, out, (unsigned int)nc, C);
  finalize_edges<<<eblocks, 256, 0, stream>>>(row, col, ew, colS, out,
                                              rowOff, colOff, wOff, E);
}